// GNN4_17471926960725
// MI455X (gfx1250) — compile-verified
//
#include <hip/hip_runtime.h>
#include <math.h>

typedef __attribute__((ext_vector_type(2))) float v2f;
typedef __attribute__((ext_vector_type(8))) float v8f;

#define N_DRUG 846
#define NS     1024
#define EDIM   64
#define WPAD   68    // drug_rel tile row stride (floats)
#define KSTR   160   // k-pair row stride for interleaved W1 (floats); 160 % 64 == 32

// ---------- pre-pass: w2sum[n][e] = sum_f W2[n][e][f] ----------
__global__ __launch_bounds__(64) void gnn_w2sum_k(const float* __restrict__ W2,
                                                  float* __restrict__ out) {
    const int n = blockIdx.x;
    const int e = threadIdx.x;
    const float* row = W2 + (size_t)n * EDIM * EDIM + (size_t)e * EDIM;
    float s = 0.f;
#pragma unroll 8
    for (int j = 0; j < EDIM; ++j) s += row[j];
    out[(size_t)n * EDIM + e] = s;
}

// ---------- pre-pass: b2sum[s] = sum_f b2[s][f] ----------
__global__ __launch_bounds__(256) void gnn_b2sum_k(const float* __restrict__ b2,
                                                   float* __restrict__ out) {
    const int s = blockIdx.x * 256 + threadIdx.x;
    const float* row = b2 + (size_t)s * EDIM;
    float a = 0.f;
#pragma unroll 8
    for (int j = 0; j < EDIM; ++j) a += row[j];
    out[s] = a;
}

// ---------- main: per-drug attention + aggregation + linear ----------
__global__ __launch_bounds__(256) void gnn_main_k(
    const int*   __restrict__ drug_name,
    const int*   __restrict__ adj_tail,
    const int*   __restrict__ adj_relation,
    const float* __restrict__ drug_table,
    const float* __restrict__ rela_table,
    const float* __restrict__ ent_table,
    const float* __restrict__ W1,
    const float* __restrict__ b1,
    const float* __restrict__ lin_W,
    const float* __restrict__ lin_b,
    const float* __restrict__ w2sumAll,
    const float* __restrict__ b2sum,
    float*       __restrict__ x_out)
{
    // W1 stored k-pair interleaved: sW1p[kp*KSTR + 2*f + c] = W1[2*kp + c][f]
    __shared__ float sW1p[32 * KSTR];         // 20.0 KB
    __shared__ float sDR[8][16 * WPAD];       // 34.8 KB per-wave drug_rel tiles
    __shared__ float sDrug[EDIM];
    __shared__ float sW2s[EDIM];
    __shared__ float sScore[NS];
    __shared__ float sRed[8];
    __shared__ float sRed2[8];
    __shared__ float sWent[EDIM];

    const int n    = blockIdx.x;
    const int t    = threadIdx.x;
    const int lane = t & 31;
    const int wid  = t >> 5;
    const int half = lane >> 4;
    const int l16  = lane & 15;

    // stage drug_emb, W1 (k-pair interleaved), w2sum into LDS
    if (t < EDIM) {
        const int dn = drug_name[n];
        sDrug[t] = drug_table[(size_t)dn * EDIM + t];
        sW2s[t]  = w2sumAll[(size_t)n * EDIM + t];
    }
    {
        const float* gW1 = W1 + (size_t)n * EDIM * EDIM;
        for (int i = t; i < EDIM * EDIM; i += 256) {
            const int k = i >> 6;
            const int f = i & 63;
            sW1p[(k >> 1) * KSTR + f * 2 + (k & 1)] = gW1[i];
        }
    }
    __syncthreads();

    // ---- per-wave 16-neighbor chunks: GEMM1 (WMMA f32) + folded GEMM2 ----
    for (int chunk = wid; chunk < NS / 16; chunk += 8) {
        const int sbase = chunk * 16;
        float* DR = &sDR[wid][0];

        // build drug_rel tile [16 x 64]: 2 lanes per row, 32 floats each
        {
            const int row = lane >> 1;
            const int hc  = lane & 1;
            const int ridx = adj_relation[(size_t)n * NS + sbase + row];
            const float* rrow = rela_table + (size_t)ridx * EDIM + hc * 32;
            float* drow = DR + row * WPAD + hc * 32;
#pragma unroll
            for (int j = 0; j < 32; j += 4) {
                float4 rv = *(const float4*)(rrow + j);
                float4 dv = *(const float4*)(&sDrug[hc * 32 + j]);
                rv.x *= dv.x; rv.y *= dv.y; rv.z *= dv.z; rv.w *= dv.w;
                *(float4*)(drow + j) = rv;
            }
        }
        asm volatile("s_wait_dscnt 0" ::: "memory");  // wave-local LDS RAW

        float part[8];
#pragma unroll
        for (int r = 0; r < 8; ++r) part[r] = 0.f;

#pragma unroll
        for (int ft = 0; ft < 4; ++ft) {
            const int f0 = ft * 16;
            // C seeded with b1 tile: row M = r + 8*half, col N = f0 + l16
            v8f c;
#pragma unroll
            for (int r = 0; r < 8; ++r)
                c[r] = b1[(size_t)(sbase + r + half * 8) * EDIM + f0 + l16];
            // K loop: 16 x V_WMMA_F32_16X16X4_F32
#pragma unroll
            for (int k0 = 0; k0 < EDIM; k0 += 4) {
                // A: rows s (lane l16), K = k0 + 2*half + {0,1}  (contiguous)
                v2f a = *(const v2f*)(DR + l16 * WPAD + k0 + half * 2);
                // B: W1[k0+2*half+{0,1}][f0+l16] — adjacent in k-pair layout
                v2f b = *(const v2f*)(sW1p + (k0 / 2 + half) * KSTR + (f0 + l16) * 2);
                c = __builtin_amdgcn_wmma_f32_16x16x4_f32(
                        false, a, false, b, (short)0, c, false, false);
            }
            // relu + folded GEMM2: score partial = sum_f relu(h) * w2sum[f]
            const float w2s = sW2s[f0 + l16];
#pragma unroll
            for (int r = 0; r < 8; ++r)
                part[r] += fmaxf(c[r], 0.f) * w2s;
        }
        // reduce across the 16 lanes of each half (cols), write scores
#pragma unroll
        for (int r = 0; r < 8; ++r) {
            float v = part[r];
            v += __shfl_xor(v, 1);
            v += __shfl_xor(v, 2);
            v += __shfl_xor(v, 4);
            v += __shfl_xor(v, 8);
            part[r] = v;
        }
        if (l16 == 0) {
#pragma unroll
            for (int r = 0; r < 8; ++r) {
                const int s = sbase + half * 8 + r;
                sScore[s] = part[r] + b2sum[s];
            }
        }
    }
    __syncthreads();

    // ---- softmax over 1024 neighbors ----
    float m = -3.4e38f;
    for (int i = t; i < NS; i += 256) m = fmaxf(m, sScore[i]);
#pragma unroll
    for (int off = 16; off >= 1; off >>= 1) m = fmaxf(m, __shfl_xor(m, off));
    if (lane == 0) sRed[wid] = m;
    __syncthreads();
    float gmax = sRed[0];
#pragma unroll
    for (int w = 1; w < 8; ++w) gmax = fmaxf(gmax, sRed[w]);

    float lsum = 0.f;
    for (int i = t; i < NS; i += 256) {
        const float ev = __expf(sScore[i] - gmax);
        sScore[i] = ev;
        lsum += ev;
    }
#pragma unroll
    for (int off = 16; off >= 1; off >>= 1) lsum += __shfl_xor(lsum, off);
    if (lane == 0) sRed2[wid] = lsum;
    if (t < EDIM) sWent[t] = 0.f;
    __syncthreads();
    float tot = 0.f;
#pragma unroll
    for (int w = 0; w < 8; ++w) tot += sRed2[w];
    const float inv = 1.0f / tot;

    // ---- weighted_ent = sum_s attn[s] * ent_table[adj_tail[n][s]] ----
    {
        float ax = 0.f, ay = 0.f;
        const int s0 = wid * 128;
        for (int s = s0; s < s0 + 128; ++s) {
            const float a = sScore[s];
            const int ti = adj_tail[(size_t)n * NS + s];
            const float2 e2 = *(const float2*)(ent_table + (size_t)ti * EDIM + lane * 2);
            ax += a * e2.x;
            ay += a * e2.y;
        }
        atomicAdd(&sWent[lane * 2 + 0], ax * inv);
        atomicAdd(&sWent[lane * 2 + 1], ay * inv);
    }
    __syncthreads();

    // ---- x = relu(lin_W @ concat(weighted_ent, drug_emb) + lin_b) ----
    if (t < EDIM) {
        const float* wrow = lin_W + (size_t)t * (2 * EDIM);
        float acc = lin_b[t];
#pragma unroll 8
        for (int j = 0; j < EDIM; ++j) acc += wrow[j] * sWent[j];
#pragma unroll 8
        for (int j = 0; j < EDIM; ++j) acc += wrow[EDIM + j] * sDrug[j];
        x_out[(size_t)n * EDIM + t] = fmaxf(acc, 0.f);
    }
}

// ---------- batchnorm over the drug dim (training-mode, biased var) ----------
__global__ __launch_bounds__(256) void gnn_bn_k(const float* __restrict__ x,
                                                const float* __restrict__ gamma,
                                                const float* __restrict__ beta,
                                                float* __restrict__ out) {
    const int f = blockIdx.x;
    const int t = threadIdx.x;
    const int lane = t & 31;
    const int wid  = t >> 5;
    __shared__ float rs[8], rs2[8];

    float v[4];
    float s = 0.f, s2 = 0.f;
#pragma unroll
    for (int i = 0; i < 4; ++i) {
        const int nn = t + i * 256;
        float val = 0.f;
        if (nn < N_DRUG) val = x[(size_t)nn * EDIM + f];
        v[i] = val;
        s  += val;
        s2 += val * val;
    }
#pragma unroll
    for (int off = 16; off >= 1; off >>= 1) {
        s  += __shfl_xor(s,  off);
        s2 += __shfl_xor(s2, off);
    }
    if (lane == 0) { rs[wid] = s; rs2[wid] = s2; }
    __syncthreads();
    float ts = 0.f, ts2 = 0.f;
#pragma unroll
    for (int w = 0; w < 8; ++w) { ts += rs[w]; ts2 += rs2[w]; }
    const float mean = ts * (1.0f / N_DRUG);
    const float var  = ts2 * (1.0f / N_DRUG) - mean * mean;
    const float sc   = rsqrtf(var + 1e-5f) * gamma[f];
    const float bb   = beta[f];
#pragma unroll
    for (int i = 0; i < 4; ++i) {
        const int nn = t + i * 256;
        if (nn < N_DRUG) out[(size_t)nn * EDIM + f] = (v[i] - mean) * sc + bb;
    }
}

extern "C" void kernel_launch(void* const* d_in, const int* in_sizes, int n_in,
                              void* d_out, int out_size, void* d_ws, size_t ws_size,
                              hipStream_t stream) {
    const int*   drug_name    = (const int*)d_in[0];
    const int*   adj_tail     = (const int*)d_in[1];
    const int*   adj_relation = (const int*)d_in[2];
    const float* drug_table   = (const float*)d_in[3];
    const float* rela_table   = (const float*)d_in[4];
    const float* ent_table    = (const float*)d_in[5];
    const float* W1           = (const float*)d_in[6];
    const float* b1           = (const float*)d_in[7];
    const float* W2           = (const float*)d_in[8];
    const float* b2           = (const float*)d_in[9];
    const float* lin_W        = (const float*)d_in[10];
    const float* lin_b        = (const float*)d_in[11];
    const float* bn_gamma     = (const float*)d_in[12];
    const float* bn_beta      = (const float*)d_in[13];

    float* ws     = (float*)d_ws;
    float* w2sum  = ws;                          // 846*64
    float* b2sum  = w2sum + N_DRUG * EDIM;       // 1024
    float* x_buf  = b2sum + NS;                  // 846*64

    gnn_w2sum_k<<<N_DRUG, 64, 0, stream>>>(W2, w2sum);
    gnn_b2sum_k<<<NS / 256, 256, 0, stream>>>(b2, b2sum);
    gnn_main_k<<<N_DRUG, 256, 0, stream>>>(drug_name, adj_tail, adj_relation,
                                           drug_table, rela_table, ent_table,
                                           W1, b1, lin_W, lin_b,
                                           w2sum, b2sum, x_buf);
    gnn_bn_k<<<EDIM, 256, 0, stream>>>(x_buf, bn_gamma, bn_beta, (float*)d_out);
}